// CoGNN_1185410973786
// MI455X (gfx1250) — compile-verified
//
#include <hip/hip_runtime.h>
#include <hip/hip_bf16.h>

#define DIM 64
#define HID 16

typedef __attribute__((ext_vector_type(2))) float v2f;
typedef __attribute__((ext_vector_type(8))) float v8f;

// ---------------------------------------------------------------- utilities
__global__ void zero_f32(float* __restrict__ p, int n) {
  int i = blockIdx.x * blockDim.x + threadIdx.x;
  if (i < n) p[i] = 0.0f;
}

__global__ void deg_kernel(const int* __restrict__ dst, float* __restrict__ deg, int E) {
  int e = blockIdx.x * blockDim.x + threadIdx.x;
  if (e < E) atomicAdd(&deg[dst[e]], 1.0f);
}

__global__ void invdeg_kernel(const float* __restrict__ deg, float* __restrict__ inv_deg, int n) {
  int i = blockIdx.x * blockDim.x + threadIdx.x;
  if (i < n) inv_deg[i] = 1.0f / fmaxf(deg[i], 1.0f);
}

// LayerNorm (no affine): one wave32 per 64-float row, float2 per lane.
__global__ __launch_bounds__(256) void ln_kernel(const float* __restrict__ h,
                                                 float* __restrict__ hn, int n) {
  int row  = blockIdx.x * 8 + (threadIdx.x >> 5);
  int lane = threadIdx.x & 31;
  if (row >= n) return;
  float2 v = ((const float2*)(h + (size_t)row * DIM))[lane];
  float s  = v.x + v.y;
  float s2 = v.x * v.x + v.y * v.y;
  #pragma unroll
  for (int o = 16; o > 0; o >>= 1) {
    s  += __shfl_xor(s,  o, 32);
    s2 += __shfl_xor(s2, o, 32);
  }
  float mu  = s * (1.0f / DIM);
  float var = s2 * (1.0f / DIM) - mu * mu;
  float rs  = rsqrtf(var + 1e-5f);
  float2 o2; o2.x = (v.x - mu) * rs; o2.y = (v.y - mu) * rs;
  ((float2*)(hn + (size_t)row * DIM))[lane] = o2;
}

// Edge-parallel scatter-sum: consecutive lanes handle consecutive features
// of one edge (coalesced gather, feature-spread atomics).
template <int F>
__global__ void scatter_sum_kernel(const float* __restrict__ h, const int* __restrict__ src,
                                   const int* __restrict__ dst, float* __restrict__ agg, int E) {
  int t = blockIdx.x * blockDim.x + threadIdx.x;
  if (t >= E * F) return;
  int e = t / F, f = t % F;
  atomicAdd(&agg[(size_t)dst[e] * F + f], h[(size_t)src[e] * F + f]);
}

template <int F>
__global__ void scale_rows_kernel(float* __restrict__ agg, const float* __restrict__ inv_deg, int n) {
  int t = blockIdx.x * blockDim.x + threadIdx.x;
  if (t >= n * F) return;
  agg[t] *= inv_deg[t / F];
}

// Weighted scatter: ew = keep_in[dst] * keep_out[src] (0/1); skip zeros.
__global__ void scatter_wsum_kernel(const float* __restrict__ h, const int* __restrict__ src,
                                    const int* __restrict__ dst, const float* __restrict__ keep_in,
                                    const float* __restrict__ keep_out, float* __restrict__ wagg,
                                    float* __restrict__ degw, int E) {
  int t = blockIdx.x * blockDim.x + threadIdx.x;
  if (t >= E * DIM) return;
  int e = t / DIM, f = t % DIM;
  int s = src[e], d = dst[e];
  float ew = keep_in[d] * keep_out[s];
  if (ew != 0.0f) {
    atomicAdd(&wagg[(size_t)d * DIM + f], ew * h[(size_t)s * DIM + f]);
    if (f == 0) atomicAdd(&degw[d], ew);
  }
}

__global__ void wnorm_kernel(float* __restrict__ wagg, const float* __restrict__ degw, int n) {
  int t = blockIdx.x * blockDim.x + threadIdx.x;
  if (t >= n * DIM) return;
  wagg[t] /= fmaxf(degw[t / DIM], 1e-6f);
}

// ---------------------------------------------------------- WMMA GEMMs (f32)
// out[n,0:64] = (relu?)(A1@W1 (+ A2@W2) + bias) (+ resid), W row-major 64x64.
// Block: 8 waves -> 2 row-tiles x 4 col-tiles = 32 rows x 64 cols.
__global__ __launch_bounds__(256) void gemm_dd_kernel(
    const float* __restrict__ A1, const float* __restrict__ W1,
    const float* __restrict__ A2, const float* __restrict__ W2,
    const float* __restrict__ bias, const float* __restrict__ resid,
    float* __restrict__ out, int n, int do_relu) {
  __shared__ float sW1[DIM * DIM];
  __shared__ float sW2[DIM * DIM];
  const int tid  = threadIdx.x;
  const bool dual = (A2 != nullptr);
  for (int i = tid; i < DIM * DIM; i += 256) {
    sW1[i] = W1[i];
    if (dual) sW2[i] = W2[i];
  }
  __syncthreads();

  const int lane    = tid & 31;
  const int wave    = tid >> 5;
  const int rowBase = blockIdx.x * 32 + (wave >> 2) * 16;
  const int m       = lane & 15;
  const int khalf   = (lane >> 4) * 2;            // K split: lanes 0-15 -> {0,1}, 16-31 -> {2,3}
  const int col     = (wave & 3) * 16 + m;
  int arow = rowBase + m; if (arow > n - 1) arow = n - 1;   // clamp (stores guarded)
  const float* a1row = A1 + (size_t)arow * DIM;

  v8f acc = {0.f, 0.f, 0.f, 0.f, 0.f, 0.f, 0.f, 0.f};
  #pragma unroll
  for (int kk = 0; kk < 16; ++kk) {
    const int k0 = kk * 4 + khalf;
    v2f a = *(const v2f*)(a1row + k0);
    v2f b; b.x = sW1[k0 * DIM + col]; b.y = sW1[(k0 + 1) * DIM + col];
    acc = __builtin_amdgcn_wmma_f32_16x16x4_f32(false, a, false, b, (short)0, acc, false, false);
  }
  if (dual) {
    const float* a2row = A2 + (size_t)arow * DIM;
    #pragma unroll
    for (int kk = 0; kk < 16; ++kk) {
      const int k0 = kk * 4 + khalf;
      v2f a = *(const v2f*)(a2row + k0);
      v2f b; b.x = sW2[k0 * DIM + col]; b.y = sW2[(k0 + 1) * DIM + col];
      acc = __builtin_amdgcn_wmma_f32_16x16x4_f32(false, a, false, b, (short)0, acc, false, false);
    }
  }
  const float bj = bias[col];
  const int rb = rowBase + (lane >> 4) * 8;       // C layout: vgpr r -> M=r (lo lanes) / r+8 (hi)
  #pragma unroll
  for (int r = 0; r < 8; ++r) {
    const int row = rb + r;
    if (row < n) {
      float v = acc[r] + bj;
      if (do_relu) v = fmaxf(v, 0.0f);
      if (resid) v += resid[(size_t)row * DIM + col];
      out[(size_t)row * DIM + col] = v;
    }
  }
}

// z[n,0:16] = relu(A1@W1 + A2@W2 + bias), W row-major 64x16. 8 waves -> 128 rows.
__global__ __launch_bounds__(256) void gemm_dh_kernel(
    const float* __restrict__ A1, const float* __restrict__ W1,
    const float* __restrict__ A2, const float* __restrict__ W2,
    const float* __restrict__ bias, float* __restrict__ out, int n) {
  __shared__ float sW1[DIM * HID];
  __shared__ float sW2[DIM * HID];
  const int tid = threadIdx.x;
  for (int i = tid; i < DIM * HID; i += 256) { sW1[i] = W1[i]; sW2[i] = W2[i]; }
  __syncthreads();

  const int lane    = tid & 31;
  const int wave    = tid >> 5;
  const int rowBase = blockIdx.x * 128 + wave * 16;
  const int m       = lane & 15;
  const int khalf   = (lane >> 4) * 2;
  int arow = rowBase + m; if (arow > n - 1) arow = n - 1;
  const float* a1row = A1 + (size_t)arow * DIM;
  const float* a2row = A2 + (size_t)arow * DIM;

  v8f acc = {0.f, 0.f, 0.f, 0.f, 0.f, 0.f, 0.f, 0.f};
  #pragma unroll
  for (int kk = 0; kk < 16; ++kk) {
    const int k0 = kk * 4 + khalf;
    v2f a = *(const v2f*)(a1row + k0);
    v2f b; b.x = sW1[k0 * HID + m]; b.y = sW1[(k0 + 1) * HID + m];
    acc = __builtin_amdgcn_wmma_f32_16x16x4_f32(false, a, false, b, (short)0, acc, false, false);
  }
  #pragma unroll
  for (int kk = 0; kk < 16; ++kk) {
    const int k0 = kk * 4 + khalf;
    v2f a = *(const v2f*)(a2row + k0);
    v2f b; b.x = sW2[k0 * HID + m]; b.y = sW2[(k0 + 1) * HID + m];
    acc = __builtin_amdgcn_wmma_f32_16x16x4_f32(false, a, false, b, (short)0, acc, false, false);
  }
  const float bj = bias[m];
  const int rb = rowBase + (lane >> 4) * 8;
  #pragma unroll
  for (int r = 0; r < 8; ++r) {
    const int row = rb + r;
    if (row < n) out[(size_t)row * HID + m] = fmaxf(acc[r] + bj, 0.0f);
  }
}

// 16->2 head + Gumbel-hard forward: keep = (l0+g0 >= l1+g1) ? 1 : 0
// (positive temperature and softmax are argmax-invariant; hard one-hot in fwd).
__global__ void keep_kernel(const float* __restrict__ z, const float* __restrict__ zagg,
                            const float* __restrict__ A2s, const float* __restrict__ A2n,
                            const float* __restrict__ A2b, const float* __restrict__ u,
                            float* __restrict__ keep, int n) {
  int i = blockIdx.x * blockDim.x + threadIdx.x;
  if (i >= n) return;
  float l0 = A2b[0], l1 = A2b[1];
  #pragma unroll
  for (int k = 0; k < HID; ++k) {
    float zv = z[(size_t)i * HID + k];
    float za = zagg[(size_t)i * HID + k];
    l0 += zv * A2s[k * 2 + 0] + za * A2n[k * 2 + 0];
    l1 += zv * A2s[k * 2 + 1] + za * A2n[k * 2 + 1];
  }
  const float eps = 1e-10f;
  float g0 = -logf(-logf(u[(size_t)i * 2 + 0] + eps) + eps);
  float g1 = -logf(-logf(u[(size_t)i * 2 + 1] + eps) + eps);
  keep[i] = ((l0 + g0) >= (l1 + g1)) ? 1.0f : 0.0f;
}

// Column mean over N rows into 64 outputs (out pre-zeroed).
__global__ __launch_bounds__(256) void colmean_kernel(const float* __restrict__ node_out,
                                                      float* __restrict__ out, int n) {
  const int f  = threadIdx.x & (DIM - 1);
  const int rl = threadIdx.x >> 6;               // 4 row-lanes per block
  const int stride = gridDim.x * 4;
  float s = 0.0f;
  for (int row = blockIdx.x * 4 + rl; row < n; row += stride)
    s += node_out[(size_t)row * DIM + f];
  atomicAdd(&out[f], s / (float)n);
}

// ---------------------------------------------------------------- launcher
extern "C" void kernel_launch(void* const* d_in, const int* in_sizes, int n_in,
                              void* d_out, int out_size, void* d_ws, size_t ws_size,
                              hipStream_t stream) {
  (void)n_in; (void)out_size; (void)ws_size;
  const float* x       = (const float*)d_in[0];
  const int*   ei      = (const int*)  d_in[1];
  const float* u_in    = (const float*)d_in[2];
  const float* u_out   = (const float*)d_in[3];
  const float* W_enc   = (const float*)d_in[4];
  const float* b_enc   = (const float*)d_in[5];
  const float* W_self  = (const float*)d_in[6];
  const float* W_nbr   = (const float*)d_in[7];
  const float* b_env   = (const float*)d_in[8];
  const float* W_dec   = (const float*)d_in[9];
  const float* b_dec   = (const float*)d_in[10];
  const float* A1_self = (const float*)d_in[11];
  const float* A1_nbr  = (const float*)d_in[12];
  const float* A1_b    = (const float*)d_in[13];
  const float* A2_self = (const float*)d_in[14];
  const float* A2_nbr  = (const float*)d_in[15];
  const float* A2_b    = (const float*)d_in[16];

  const int N = in_sizes[0] / DIM;
  const int E = in_sizes[1] / 2;
  const int L = in_sizes[6] / (DIM * DIM);
  const int* src = ei;
  const int* dst = ei + E;

  // Workspace layout (~46 MB of f32)
  float* ws       = (float*)d_ws;
  float* h        = ws;  ws += (size_t)N * DIM;
  float* hn       = ws;  ws += (size_t)N * DIM;
  float* agg      = ws;  ws += (size_t)N * DIM;   // mean-agg / wmean-agg / node_out
  float* z        = ws;  ws += (size_t)N * HID;
  float* zagg     = ws;  ws += (size_t)N * HID;
  float* keep_in  = ws;  ws += N;
  float* keep_out = ws;  ws += N;
  float* deg      = ws;  ws += N;
  float* inv_deg  = ws;  ws += N;
  float* degw     = ws;  ws += N;

  const int TB = 256;
  auto cdiv = [](long long a, long long b) { return (int)((a + b - 1) / b); };

  // degrees
  zero_f32<<<cdiv(N, TB), TB, 0, stream>>>(deg, N);
  deg_kernel<<<cdiv(E, TB), TB, 0, stream>>>(dst, deg, E);
  invdeg_kernel<<<cdiv(N, TB), TB, 0, stream>>>(deg, inv_deg, N);

  // encoder: h = x @ W_enc + b_enc
  gemm_dd_kernel<<<cdiv(N, 32), TB, 0, stream>>>(x, W_enc, nullptr, nullptr, b_enc, nullptr, h, N, 0);

  for (int l = 0; l < L; ++l) {
    ln_kernel<<<cdiv(N, 8), TB, 0, stream>>>(h, hn, N);

    // mean_agg(hn)
    zero_f32<<<cdiv((long long)N * DIM, TB), TB, 0, stream>>>(agg, N * DIM);
    scatter_sum_kernel<DIM><<<cdiv((long long)E * DIM, TB), TB, 0, stream>>>(hn, src, dst, agg, E);
    scale_rows_kernel<DIM><<<cdiv((long long)N * DIM, TB), TB, 0, stream>>>(agg, inv_deg, N);

    for (int i = 0; i < 2; ++i) {
      gemm_dh_kernel<<<cdiv(N, 128), TB, 0, stream>>>(hn, A1_self + i * DIM * HID,
                                                      agg, A1_nbr + i * DIM * HID,
                                                      A1_b + i * HID, z, N);
      zero_f32<<<cdiv((long long)N * HID, TB), TB, 0, stream>>>(zagg, N * HID);
      scatter_sum_kernel<HID><<<cdiv((long long)E * HID, TB), TB, 0, stream>>>(z, src, dst, zagg, E);
      scale_rows_kernel<HID><<<cdiv((long long)N * HID, TB), TB, 0, stream>>>(zagg, inv_deg, N);
      keep_kernel<<<cdiv(N, TB), TB, 0, stream>>>(z, zagg, A2_self + i * HID * 2,
                                                  A2_nbr + i * HID * 2, A2_b + i * 2,
                                                  (i == 0 ? u_in : u_out) + (size_t)l * N * 2,
                                                  (i == 0 ? keep_in : keep_out), N);
    }

    // weighted mean aggregation (reuse agg)
    zero_f32<<<cdiv((long long)N * DIM, TB), TB, 0, stream>>>(agg, N * DIM);
    zero_f32<<<cdiv(N, TB), TB, 0, stream>>>(degw, N);
    scatter_wsum_kernel<<<cdiv((long long)E * DIM, TB), TB, 0, stream>>>(hn, src, dst, keep_in,
                                                                         keep_out, agg, degw, E);
    wnorm_kernel<<<cdiv((long long)N * DIM, TB), TB, 0, stream>>>(agg, degw, N);

    // h = relu(hn@W_self[l] + wmean@W_nbr[l] + b_env[l]) + h  (in-place residual)
    gemm_dd_kernel<<<cdiv(N, 32), TB, 0, stream>>>(hn, W_self + (size_t)l * DIM * DIM,
                                                   agg, W_nbr + (size_t)l * DIM * DIM,
                                                   b_env + (size_t)l * DIM, h, h, N, 1);
  }

  // decode + column mean
  ln_kernel<<<cdiv(N, 8), TB, 0, stream>>>(h, hn, N);
  gemm_dd_kernel<<<cdiv(N, 32), TB, 0, stream>>>(hn, W_dec, nullptr, nullptr, b_dec, nullptr, agg, N, 0);
  zero_f32<<<1, 64, 0, stream>>>((float*)d_out, DIM);
  colmean_kernel<<<512, TB, 0, stream>>>(agg, (float*)d_out, N);
}